// FEBw_32057635897472
// MI455X (gfx1250) — compile-verified
//
#include <hip/hip_runtime.h>

// Problem constants (from reference): B=8, T=4096, H*D=512, C=64, K=8, CK=512, MODES=32, L=2
#define NB      8
#define SEQ     4096
#define CKD     512
#define NMODES  32

typedef float v2f __attribute__((ext_vector_type(2)));
typedef float v8f __attribute__((ext_vector_type(8)));

// ---------------------------------------------------------------------------
// FP32 WMMA GEMM (v_wmma_f32_16x16x4_f32), LDS-tiled, batched via blockIdx.z.
// LDS layouts: As[m][k], Bs[n][k]  (k contiguous for both -> every fragment is
// one aligned 8-byte LDS read, no cross-VGPR repacking in the inner loop).
// LOADMODE: 0 = A row-major (MxK),  B row-major (KxN)
//           1 = A stored   (KxM),  B row-major (KxN)   (i.e. C = A^T B)
//           2 = A row-major (MxK),  B stored   (NxK)   (i.e. C = A B^T)
// ---------------------------------------------------------------------------
template <int BM, int BN, int BK, int LOADMODE, bool BIAS>
__global__ __launch_bounds__((BM / 16) * (BN / 64) * 32)
void wmma_gemm(const float* __restrict__ Ag, const float* __restrict__ Bg,
               const float* __restrict__ bias, float* __restrict__ Cg,
               int M, int N, int Kdim, long strideA, long strideB, long strideC)
{
    constexpr int WROWS   = BM / 16;
    constexpr int WCOLS   = BN / 64;
    constexpr int THREADS = WROWS * WCOLS * 32;

    __shared__ float As[BM][BK + 4];   // row stride 36 floats = 144B (8B aligned)
    __shared__ float Bs[BN][BK + 4];   // transposed: k contiguous

    Ag += (long)blockIdx.z * strideA;
    Bg += (long)blockIdx.z * strideB;
    Cg += (long)blockIdx.z * strideC;

    const int m0 = blockIdx.x * BM;
    const int n0 = blockIdx.y * BN;

    const int wave = threadIdx.x >> 5;
    const int lane = threadIdx.x & 31;
    const int wr   = wave % WROWS;
    const int wc   = wave / WROWS;
    const int mb   = wr * 16;
    const int nb   = wc * 64;
    const int lh   = lane >> 4;   // lane-half: selects K pair {0,1} vs {2,3}
    const int l16  = lane & 15;

    v8f acc[4] = {};

    for (int k0 = 0; k0 < Kdim; k0 += BK) {
        // ---- stage A tile into LDS as As[m][k] ----
        if (LOADMODE == 1) {                       // A stored (K x M), lda = M
            constexpr int VM = BM / 4;
            for (int idx = threadIdx.x; idx < BK * VM; idx += THREADS) {
                const int k  = idx / VM;
                const int mq = idx % VM;
                const float4 v = *(const float4*)(Ag + (long)(k0 + k) * M + (m0 + 4 * mq));
                As[4 * mq + 0][k] = v.x;  As[4 * mq + 1][k] = v.y;
                As[4 * mq + 2][k] = v.z;  As[4 * mq + 3][k] = v.w;
            }
        } else {                                   // A row-major (M x K)
            constexpr int VK = BK / 4;
            for (int idx = threadIdx.x; idx < BM * VK; idx += THREADS) {
                const int m  = idx / VK;
                const int kq = idx % VK;
                *(float4*)&As[m][4 * kq] =
                    *(const float4*)(Ag + (long)(m0 + m) * Kdim + (k0 + 4 * kq));
            }
        }
        // ---- stage B tile into LDS as Bs[n][k] ----
        if (LOADMODE == 2) {                       // B stored (N x K): direct copy
            constexpr int VK = BK / 4;
            for (int idx = threadIdx.x; idx < BN * VK; idx += THREADS) {
                const int n  = idx / VK;
                const int kq = idx % VK;
                *(float4*)&Bs[n][4 * kq] =
                    *(const float4*)(Bg + (long)(n0 + n) * Kdim + (k0 + 4 * kq));
            }
        } else {                                   // B row-major (K x N): transpose
            constexpr int VN = BN / 4;
            for (int idx = threadIdx.x; idx < BK * VN; idx += THREADS) {
                const int k  = idx / VN;
                const int nq = idx % VN;
                const float4 v = *(const float4*)(Bg + (long)(k0 + k) * N + (n0 + 4 * nq));
                Bs[4 * nq + 0][k] = v.x;  Bs[4 * nq + 1][k] = v.y;
                Bs[4 * nq + 2][k] = v.z;  Bs[4 * nq + 3][k] = v.w;
            }
        }
        __syncthreads();

        if (k0 + BK < Kdim) {   // pull next tiles toward the caches
            __builtin_prefetch(Ag + (LOADMODE == 1 ? (long)(k0 + BK) * M + m0
                                                   : (long)m0 * Kdim + k0 + BK), 0, 1);
            __builtin_prefetch(Bg + (LOADMODE == 2 ? (long)n0 * Kdim + k0 + BK
                                                   : (long)(k0 + BK) * N + n0), 0, 1);
        }

        #pragma unroll
        for (int kk = 0; kk < BK; kk += 4) {
            const int ka = kk + 2 * lh;            // even -> 8B-aligned LDS reads
            // A frag (16x4 f32): lanes 0-15 hold K={kk,kk+1}, lanes 16-31 K={kk+2,kk+3}
            const v2f af = *(const v2f*)&As[mb + l16][ka];
            #pragma unroll
            for (int t = 0; t < 4; ++t) {
                const int n = nb + t * 16 + l16;
                // B frag (4x16 f32): VGPR0 rows {kk,kk+2}, VGPR1 rows {kk+1,kk+3}
                const v2f bf = *(const v2f*)&Bs[n][ka];
                acc[t] = __builtin_amdgcn_wmma_f32_16x16x4_f32(
                    false, af, false, bf, (short)0, acc[t], false, false);
            }
        }
        __syncthreads();
    }

    // C/D layout: VGPR j -> row j (lanes 0-15) / row j+8 (lanes 16-31), col = lane&15
    #pragma unroll
    for (int t = 0; t < 4; ++t) {
        const int n  = n0 + nb + t * 16 + l16;
        const float bv = BIAS ? bias[n] : 0.f;
        #pragma unroll
        for (int j = 0; j < 8; ++j) {
            const int m = m0 + mb + j + 8 * lh;
            Cg[(long)m * N + n] = acc[t][j] + bv;
        }
    }
}

// ---------------------------------------------------------------------------
// Truncated-(i)DFT basis tables.
// forward:  E[t][2x] = cos(2pi x t / n),  E[t][2x+1] = -sin(2pi x t / n)
// inverse:  same columns scaled by 1/n (x=0) or 2/n (x>0)   [irfft with only
//           32 live modes; Im of DC is ignored since sin(0)=0]
// ---------------------------------------------------------------------------
__global__ void fill_basis(float* __restrict__ E, int n, int inverse)
{
    const int idx = blockIdx.x * 256 + threadIdx.x;
    if (idx >= n * NMODES) return;
    const int t = idx >> 5;
    const int x = idx & 31;
    const int r = (int)(((long)x * t) % n);          // exact phase reduction
    const float ang = -6.283185307179586f * (float)r / (float)n;
    const float sc  = inverse ? ((x == 0 ? 1.f : 2.f) / (float)n) : 1.f;
    E[(long)t * 64 + 2 * x]     = sc * cosf(ang);
    E[(long)t * 64 + 2 * x + 1] = sc * sinf(ang);
}

// ---------------------------------------------------------------------------
// Even/odd split + encode: xa = [cur[2t], cur[2t+1]] (16), d = xa@ec_d, s = xa@ec_s
// One thread per (b, t2, c); cur is (B, 2*half_n, 512).
// ---------------------------------------------------------------------------
__global__ void split_encode(const float* __restrict__ cur,
                             const float* __restrict__ ecd, const float* __restrict__ ecs,
                             float* __restrict__ dcoef, float* __restrict__ scoef,
                             int half_n)
{
    __shared__ float wd[16][8], ws_[16][8];
    if (threadIdx.x < 128)      wd[threadIdx.x >> 3][threadIdx.x & 7]  = ecd[threadIdx.x];
    else if (threadIdx.x < 256) { int i = threadIdx.x - 128; ws_[i >> 3][i & 7] = ecs[i]; }
    __syncthreads();

    const int idx = blockIdx.x * 256 + threadIdx.x;
    const int total = NB * half_n * 64;
    if (idx >= total) return;
    const int b  = idx / (half_n * 64);
    const int t2 = (idx / 64) % half_n;
    const int c  = idx & 63;

    const long e0 = ((long)b * 2 * half_n + 2 * t2) * CKD + c * 8;
    float a[16];
    *(float4*)&a[0]  = *(const float4*)(cur + e0);
    *(float4*)&a[4]  = *(const float4*)(cur + e0 + 4);
    *(float4*)&a[8]  = *(const float4*)(cur + e0 + CKD);
    *(float4*)&a[12] = *(const float4*)(cur + e0 + CKD + 4);

    float d[8] = {}, s[8] = {};
    #pragma unroll
    for (int j = 0; j < 16; ++j)
        #pragma unroll
        for (int k = 0; k < 8; ++k) {
            d[k] = fmaf(a[j], wd[j][k], d[k]);
            s[k] = fmaf(a[j], ws_[j][k], s[k]);
        }

    const long o0 = ((long)b * half_n + t2) * CKD + c * 8;
    *(float4*)(dcoef + o0)     = *(float4*)&d[0];
    *(float4*)(dcoef + o0 + 4) = *(float4*)&d[4];
    *(float4*)(scoef + o0)     = *(float4*)&s[0];
    *(float4*)(scoef + o0 + 4) = *(float4*)&s[4];
}

// ---------------------------------------------------------------------------
// Per-mode complex channel mixing, fused over the three spectral weights:
//   UdF[b,o,x] = sum_i Df[b,i,x]*wA[i,o,x] + Sf[b,i,x]*wB[i,o,x]
//   UsF[b,o,x] = sum_i Df[b,i,x]*wC[i,o,x]
// Layouts: Df/Sf/UdF/UsF = [b][ch][64] (re,im interleaved); w = [i][o][x].
// threadIdx.x&31 = mode x  ->  coalesced weight streaming (x is innermost).
// ---------------------------------------------------------------------------
#define MIX_CHUNK 4
__global__ void mode_mix(const float* __restrict__ Df, const float* __restrict__ Sf,
                         const float* __restrict__ wAre, const float* __restrict__ wAim,
                         const float* __restrict__ wBre, const float* __restrict__ wBim,
                         const float* __restrict__ wCre, const float* __restrict__ wCim,
                         float* __restrict__ UdF, float* __restrict__ UsF)
{
    const int x  = threadIdx.x & 31;
    const int ol = threadIdx.x >> 5;            // 0..7
    const int o  = blockIdx.x * 8 + ol;

    __shared__ float sD[MIX_CHUNK][NB][64];
    __shared__ float sS[MIX_CHUNK][NB][64];

    float udr[NB] = {}, udi[NB] = {}, usr[NB] = {}, usi[NB] = {};

    for (int i0 = 0; i0 < CKD; i0 += MIX_CHUNK) {
        for (int t = threadIdx.x; t < MIX_CHUNK * NB * 64; t += 256) {
            const int ii = t / (NB * 64);
            const int b  = (t / 64) % NB;
            const int f  = t & 63;
            const long src = ((long)b * CKD + (i0 + ii)) * 64 + f;
            sD[ii][b][f] = Df[src];
            sS[ii][b][f] = Sf[src];
        }
        __syncthreads();

        #pragma unroll
        for (int ii = 0; ii < MIX_CHUNK; ++ii) {
            const long wb = ((long)(i0 + ii) * CKD + o) * NMODES + x;
            const float ar = wAre[wb], ai = wAim[wb];
            const float br = wBre[wb], bi = wBim[wb];
            const float cr = wCre[wb], ci = wCim[wb];
            #pragma unroll
            for (int b = 0; b < NB; ++b) {
                const float dr = sD[ii][b][2 * x], di = sD[ii][b][2 * x + 1];
                const float sr = sS[ii][b][2 * x], si = sS[ii][b][2 * x + 1];
                udr[b] += dr * ar - di * ai + sr * br - si * bi;
                udi[b] += dr * ai + di * ar + sr * bi + si * br;
                usr[b] += dr * cr - di * ci;
                usi[b] += dr * ci + di * cr;
            }
        }
        __syncthreads();
    }

    #pragma unroll
    for (int b = 0; b < NB; ++b) {
        const long ob = ((long)b * CKD + o) * 64 + 2 * x;
        UdF[ob] = udr[b];  UdF[ob + 1] = udi[b];
        UsF[ob] = usr[b];  UsF[ob + 1] = usi[b];
    }
}

// ---------------------------------------------------------------------------
// cur @ t0_w + t0_b over the last (K=8) axis. One thread per (b,t,c).
// ---------------------------------------------------------------------------
__global__ void apply_t0(const float* __restrict__ in, const float* __restrict__ t0w,
                         const float* __restrict__ t0b, float* __restrict__ out, int total)
{
    __shared__ float w[8][8], bb[8];
    if (threadIdx.x < 64) w[threadIdx.x >> 3][threadIdx.x & 7] = t0w[threadIdx.x];
    if (threadIdx.x < 8)  bb[threadIdx.x] = t0b[threadIdx.x];
    __syncthreads();

    const int idx = blockIdx.x * 256 + threadIdx.x;
    if (idx >= total) return;
    const long base = (long)idx * 8;
    float v[8];
    *(float4*)&v[0] = *(const float4*)(in + base);
    *(float4*)&v[4] = *(const float4*)(in + base + 4);
    float r[8];
    #pragma unroll
    for (int k = 0; k < 8; ++k) r[k] = bb[k];
    #pragma unroll
    for (int j = 0; j < 8; ++j)
        #pragma unroll
        for (int k = 0; k < 8; ++k) r[k] = fmaf(v[j], w[j][k], r[k]);
    *(float4*)(out + base)     = *(float4*)&r[0];
    *(float4*)(out + base + 4) = *(float4*)&r[4];
}

// ---------------------------------------------------------------------------
// Inverse wavelet step: v = cur + Us;  [v, Ud] @ rc_e -> even row, @ rc_o -> odd.
// One thread per (b, t, c); output has 2*half_n rows.
// ---------------------------------------------------------------------------
__global__ void recon(const float* __restrict__ cur, const float* __restrict__ us,
                      const float* __restrict__ ud,
                      const float* __restrict__ rce, const float* __restrict__ rco,
                      float* __restrict__ out, int half_n)
{
    __shared__ float we[16][8], wo[16][8];
    if (threadIdx.x < 128)      we[threadIdx.x >> 3][threadIdx.x & 7] = rce[threadIdx.x];
    else if (threadIdx.x < 256) { int i = threadIdx.x - 128; wo[i >> 3][i & 7] = rco[i]; }
    __syncthreads();

    const int idx = blockIdx.x * 256 + threadIdx.x;
    const int total = NB * half_n * 64;
    if (idx >= total) return;
    const int b = idx / (half_n * 64);
    const int t = (idx / 64) % half_n;
    const int c = idx & 63;

    const long ib = ((long)b * half_n + t) * CKD + c * 8;
    float a[16];
    #pragma unroll
    for (int j = 0; j < 8; ++j) a[j] = cur[ib + j] + us[ib + j];
    #pragma unroll
    for (int j = 0; j < 8; ++j) a[8 + j] = ud[ib + j];

    float e[8] = {}, o[8] = {};
    #pragma unroll
    for (int j = 0; j < 16; ++j)
        #pragma unroll
        for (int k = 0; k < 8; ++k) {
            e[k] = fmaf(a[j], we[j][k], e[k]);
            o[k] = fmaf(a[j], wo[j][k], o[k]);
        }

    const long ob = ((long)b * 2 * half_n + 2 * t) * CKD + c * 8;
    *(float4*)(out + ob)           = *(float4*)&e[0];
    *(float4*)(out + ob + 4)       = *(float4*)&e[4];
    *(float4*)(out + ob + CKD)     = *(float4*)&o[0];
    *(float4*)(out + ob + CKD + 4) = *(float4*)&o[4];
}

// ---------------------------------------------------------------------------
extern "C" void kernel_launch(void* const* d_in, const int* in_sizes, int n_in,
                              void* d_out, int out_size, void* d_ws, size_t ws_size,
                              hipStream_t stream)
{
    const float* x         = (const float*)d_in[0];
    const float* lin_in_w  = (const float*)d_in[1];
    const float* lin_in_b  = (const float*)d_in[2];
    const float* lin_out_w = (const float*)d_in[3];
    const float* lin_out_b = (const float*)d_in[4];
    const float* t0_w      = (const float*)d_in[5];
    const float* t0_b      = (const float*)d_in[6];
    const float* wA_re     = (const float*)d_in[7];
    const float* wA_im     = (const float*)d_in[8];
    const float* wB_re     = (const float*)d_in[9];
    const float* wB_im     = (const float*)d_in[10];
    const float* wC_re     = (const float*)d_in[11];
    const float* wC_im     = (const float*)d_in[12];
    const float* ec_s      = (const float*)d_in[13];
    const float* ec_d      = (const float*)d_in[14];
    const float* rc_e      = (const float*)d_in[15];
    const float* rc_o      = (const float*)d_in[16];

    float* ws = (float*)d_ws;
    size_t off = 0;
    auto alloc = [&](size_t n) { float* p = ws + off; off += n; return p; };

    float* EB0 = alloc(2048 * 64);               // forward basis, n=2048
    float* EB1 = alloc(1024 * 64);               // forward basis, n=1024
    float* RB0 = alloc(2048 * 64);               // inverse basis, n=2048
    float* RB1 = alloc(1024 * 64);               // inverse basis, n=1024
    float* z0  = alloc((size_t)NB * 4096 * CKD); // lin_in out; reused as cur(4096)
    float* d0  = alloc((size_t)NB * 2048 * CKD); // level-0 dcoef; reused as cur(2048)
    float* s0  = alloc((size_t)NB * 2048 * CKD);
    float* Ud0 = alloc((size_t)NB * 2048 * CKD);
    float* Us0 = alloc((size_t)NB * 2048 * CKD);
    float* d1  = alloc((size_t)NB * 1024 * CKD);
    float* s1  = alloc((size_t)NB * 1024 * CKD);
    float* Ud1 = alloc((size_t)NB * 1024 * CKD);
    float* Us1 = alloc((size_t)NB * 1024 * CKD);
    float* c1  = alloc((size_t)NB * 1024 * CKD); // after t0
    float* Df  = alloc((size_t)NB * CKD * 64);
    float* Sf  = alloc((size_t)NB * CKD * 64);
    float* UdF = alloc((size_t)NB * CKD * 64);
    float* UsF = alloc((size_t)NB * CKD * 64);

    // --- basis tables ---
    fill_basis<<<(2048 * 32 + 255) / 256, 256, 0, stream>>>(EB0, 2048, 0);
    fill_basis<<<(1024 * 32 + 255) / 256, 256, 0, stream>>>(EB1, 1024, 0);
    fill_basis<<<(2048 * 32 + 255) / 256, 256, 0, stream>>>(RB0, 2048, 1);
    fill_basis<<<(1024 * 32 + 255) / 256, 256, 0, stream>>>(RB1, 1024, 1);

    const int M0 = NB * SEQ;                      // 32768 tokens

    // --- lin_in: z0 = x @ W + b ---
    wmma_gemm<64, 128, 32, 0, true><<<dim3(M0 / 64, CKD / 128, 1), 256, 0, stream>>>(
        x, lin_in_w, lin_in_b, z0, M0, CKD, CKD, 0, 0, 0);

    // ================= level 0 (n = 4096 -> half 2048) =================
    split_encode<<<(NB * 2048 * 64 + 255) / 256, 256, 0, stream>>>(z0, ec_d, ec_s, d0, s0, 2048);

    {   // truncated DFT: Freq(512x64) = coef^T @ E   (TN GEMM, K = 2048)
        dim3 g(CKD / 64, 1, NB);
        wmma_gemm<64, 64, 32, 1, false><<<g, 128, 0, stream>>>(
            d0, EB0, nullptr, Df, CKD, 64, 2048, (long)2048 * CKD, 0, (long)CKD * 64);
        wmma_gemm<64, 64, 32, 1, false><<<g, 128, 0, stream>>>(
            s0, EB0, nullptr, Sf, CKD, 64, 2048, (long)2048 * CKD, 0, (long)CKD * 64);
    }
    mode_mix<<<64, 256, 0, stream>>>(Df, Sf, wA_re, wA_im, wB_re, wB_im, wC_re, wC_im, UdF, UsF);
    {   // truncated iDFT: Y(2048x512) = R @ G^T   (NT GEMM, K = 64)
        dim3 g(2048 / 64, CKD / 128, NB);
        wmma_gemm<64, 128, 32, 2, false><<<g, 256, 0, stream>>>(
            RB0, UdF, nullptr, Ud0, 2048, CKD, 64, 0, (long)CKD * 64, (long)2048 * CKD);
        wmma_gemm<64, 128, 32, 2, false><<<g, 256, 0, stream>>>(
            RB0, UsF, nullptr, Us0, 2048, CKD, 64, 0, (long)CKD * 64, (long)2048 * CKD);
    }

    // ================= level 1 (n = 2048 -> half 1024) =================
    split_encode<<<(NB * 1024 * 64 + 255) / 256, 256, 0, stream>>>(s0, ec_d, ec_s, d1, s1, 1024);
    {
        dim3 g(CKD / 64, 1, NB);
        wmma_gemm<64, 64, 32, 1, false><<<g, 128, 0, stream>>>(
            d1, EB1, nullptr, Df, CKD, 64, 1024, (long)1024 * CKD, 0, (long)CKD * 64);
        wmma_gemm<64, 64, 32, 1, false><<<g, 128, 0, stream>>>(
            s1, EB1, nullptr, Sf, CKD, 64, 1024, (long)1024 * CKD, 0, (long)CKD * 64);
    }
    mode_mix<<<64, 256, 0, stream>>>(Df, Sf, wA_re, wA_im, wB_re, wB_im, wC_re, wC_im, UdF, UsF);
    {
        dim3 g(1024 / 64, CKD / 128, NB);
        wmma_gemm<64, 128, 32, 2, false><<<g, 256, 0, stream>>>(
            RB1, UdF, nullptr, Ud1, 1024, CKD, 64, 0, (long)CKD * 64, (long)1024 * CKD);
        wmma_gemm<64, 128, 32, 2, false><<<g, 256, 0, stream>>>(
            RB1, UsF, nullptr, Us1, 1024, CKD, 64, 0, (long)CKD * 64, (long)1024 * CKD);
    }

    // ================= coarsest transform + reconstruction =================
    apply_t0<<<(NB * 1024 * 64 + 255) / 256, 256, 0, stream>>>(s1, t0_w, t0_b, c1, NB * 1024 * 64);

    // level-1 inverse -> cur(2048) (reuse d0)
    recon<<<(NB * 1024 * 64 + 255) / 256, 256, 0, stream>>>(c1, Us1, Ud1, rc_e, rc_o, d0, 1024);
    // level-0 inverse -> cur(4096) (reuse z0)
    recon<<<(NB * 2048 * 64 + 255) / 256, 256, 0, stream>>>(d0, Us0, Ud0, rc_e, rc_o, z0, 2048);

    // --- lin_out: out = cur @ W + b ---
    wmma_gemm<64, 128, 32, 0, true><<<dim3(M0 / 64, CKD / 128, 1), 256, 0, stream>>>(
        z0, lin_out_w, lin_out_b, (float*)d_out, M0, CKD, CKD, 0, 0, 0);
}